// ParametricKernel_85564338471564
// MI455X (gfx1250) — compile-verified
//
#include <hip/hip_runtime.h>
#include <hip/hip_bf16.h>

#define MDIM 64
#define LDSS 65   // padded LDS stride to avoid bank conflicts on column access

typedef float v2f __attribute__((ext_vector_type(2)));
typedef float v8f __attribute__((ext_vector_type(8)));

__device__ __forceinline__ float matern52(float d2, float inv_ls) {
    // (1 + sqrt5*r + 5/3 r^2) * exp(-sqrt5*r), r = sqrt(max(d2,1e-30))/ls
    float r = sqrtf(fmaxf(d2, 1e-30f)) * inv_ls;
    float s = 2.23606797749979f * r;
    return (1.0f + s + 1.6666666666666667f * r * r) * __expf(-s);
}

// One wave (32 threads) per row. Lane owns matrix rows (lane) and (lane+32).
// Blocked Cholesky: 16-wide panels factored scalar-lockstep, trailing SYRK
// updates done with chained V_WMMA_F32_16X16X4_F32 (f32-exact, no cancellation).
__global__ __launch_bounds__(32) void vecchia_solve_kernel(
    const int* __restrict__ batch_idx, const float* __restrict__ locs,
    const int* __restrict__ nn, const float* __restrict__ log_ls,
    float* __restrict__ out, int N)
{
    __shared__ float S[MDIM * LDSS];
    __shared__ float px[MDIM], py[MDIM], bv[MDIM];

    const int lane = threadIdx.x;          // 0..31
    const int r0 = lane, r1 = lane + 32;
    const int half = lane >> 4;            // 0: lanes 0-15, 1: lanes 16-31
    const int l    = lane & 15;
    const float inv_ls = __expf(-log_ls[0]);   // 1/ls = exp(-log_ls)

    for (int row = blockIdx.x; row < N; row += gridDim.x) {
        const int ptr = batch_idx[row];
        const int nv  = (ptr < MDIM) ? ptr : MDIM;   // valid prefix length

        // ---- gather neighbor coordinates ----
        {
            int i0 = nn[ptr * MDIM + r0];
            int i1 = nn[ptr * MDIM + r1];
            px[r0] = locs[2 * i0];  py[r0] = locs[2 * i0 + 1];
            px[r1] = locs[2 * i1];  py[r1] = locs[2 * i1 + 1];
        }
        __syncthreads();

        // ---- build lower triangle of S22 and rhs S12 (direct distances) ----
        const float cx = locs[2 * ptr], cy = locs[2 * ptr + 1];
        #pragma unroll
        for (int rr = 0; rr < 2; ++rr) {
            const int r = rr ? r1 : r0;
            const float xr = px[r], yr = py[r];
            if (r < nv) {
                for (int c = 0; c < r; ++c) {
                    float dx = xr - px[c], dy = yr - py[c];
                    S[r * LDSS + c] = matern52(dx * dx + dy * dy, inv_ls);
                }
                S[r * LDSS + r] = 1.0f + 1e-12f;   // matern(0) + nugget^2
                float dx = xr - cx, dy = yr - cy;
                bv[r] = matern52(dx * dx + dy * dy, inv_ls);
            } else {
                for (int c = 0; c < r; ++c) S[r * LDSS + c] = 0.0f;
                S[r * LDSS + r] = 1.0f;            // identity padding
                bv[r] = 0.0f;
            }
        }
        __syncthreads();

        // ---- blocked Cholesky (lower), panels of 16 columns ----
        for (int kb = 0; kb < 4; ++kb) {
            const int j0 = kb * 16;
            // panel factorization (scalar lockstep, updates confined to panel)
            for (int jj = 0; jj < 16; ++jj) {
                const int j = j0 + jj;
                float d   = sqrtf(S[j * LDSS + j]);
                float inv = 1.0f / d;
                if (lane == 0) S[j * LDSS + j] = d;
                if (r0 > j) S[r0 * LDSS + j] *= inv;
                if (r1 > j) S[r1 * LDSS + j] *= inv;
                __syncthreads();
                #pragma unroll
                for (int rr = 0; rr < 2; ++rr) {
                    const int r = rr ? r1 : r0;
                    if (r > j) {
                        float ljk = S[r * LDSS + j];
                        int ce = j0 + 15; if (r < ce) ce = r;
                        for (int c = j + 1; c <= ce; ++c)
                            S[r * LDSS + c] -= ljk * S[c * LDSS + j];
                    }
                }
                __syncthreads();
            }
            // trailing update: tile(tr,tc) -= Lpanel[tr] * Lpanel[tc]^T
            for (int tr = kb + 1; tr < 4; ++tr) {
                for (int tc = kb + 1; tc <= tr; ++tc) {
                    v8f c;
                    #pragma unroll
                    for (int vi = 0; vi < 8; ++vi)
                        c[vi] = S[(tr * 16 + half * 8 + vi) * LDSS + tc * 16 + l];
                    #pragma unroll
                    for (int q = 0; q < 4; ++q) {
                        const int k0 = j0 + 4 * q + half * 2;  // lanes16-31 supply K2,K3
                        v2f a, b;
                        a[0] = -S[(tr * 16 + l) * LDSS + k0];      // negate A => C - L*L^T
                        a[1] = -S[(tr * 16 + l) * LDSS + k0 + 1];
                        b[0] =  S[(tc * 16 + l) * LDSS + k0];
                        b[1] =  S[(tc * 16 + l) * LDSS + k0 + 1];
                        c = __builtin_amdgcn_wmma_f32_16x16x4_f32(
                                false, a, false, b, (short)0, c, false, false);
                    }
                    #pragma unroll
                    for (int vi = 0; vi < 8; ++vi)
                        S[(tr * 16 + half * 8 + vi) * LDSS + tc * 16 + l] = c[vi];
                }
            }
            __syncthreads();
        }

        // ---- forward solve  L y = b  (y kept in bv) ----
        for (int k = 0; k < MDIM; ++k) {
            float vk = bv[k] / S[k * LDSS + k];
            if (lane == 0) bv[k] = vk;
            if (r0 > k) bv[r0] -= S[r0 * LDSS + k] * vk;
            if (r1 > k) bv[r1] -= S[r1 * LDSS + k] * vk;
            __syncthreads();
        }
        // ---- backward solve  L^T w = y ----
        for (int k = MDIM - 1; k >= 0; --k) {
            float wk = bv[k] / S[k * LDSS + k];
            if (lane == 0) bv[k] = wk;
            if (r0 < k) bv[r0] -= S[k * LDSS + r0] * wk;
            if (r1 < k) bv[r1] -= S[k * LDSS + r1] * wk;
            __syncthreads();
        }

        out[row * MDIM + r0] = bv[r0];
        out[row * MDIM + r1] = bv[r1];
        __syncthreads();
    }
}

// Deterministic column reduction: avg_j = (sum_i w[i][j]^2) / (#{i: batch_idx[i] > j}),
// m = #{j : avg_j >= 1e-4}. One block, fixed partition + tree reduce (no float atomics).
__global__ __launch_bounds__(1024) void vecchia_reduce_kernel(
    const float* __restrict__ out, const int* __restrict__ batch_idx,
    int N, int* __restrict__ m_out)
{
    __shared__ float ssum[1024];
    __shared__ int   scnt[1024];
    __shared__ int   mcnt;
    const int tid = threadIdx.x;
    const int col = tid & 63;
    const int g   = tid >> 6;        // 16 row-groups
    float s = 0.0f; int c = 0;
    for (int i = g; i < N; i += 16) {
        float w = out[i * MDIM + col];
        s += w * w;
        c += (batch_idx[i] > col) ? 1 : 0;
    }
    ssum[tid] = s; scnt[tid] = c;
    if (tid == 0) mcnt = 0;
    __syncthreads();
    for (int off = 512; off >= 64; off >>= 1) {
        if (tid < off) { ssum[tid] += ssum[tid + off]; scnt[tid] += scnt[tid + off]; }
        __syncthreads();
    }
    if (tid < 64) {
        float avg = ssum[tid] / (float)scnt[tid];
        if (avg >= 1e-4f) atomicAdd(&mcnt, 1);
    }
    __syncthreads();
    if (tid == 0) *m_out = mcnt;
}

// Zero columns >= m, write variances = 1.
__global__ void vecchia_mask_kernel(float* __restrict__ out,
                                    const int* __restrict__ m_ptr, int N)
{
    const int idx = blockIdx.x * blockDim.x + threadIdx.x;
    const int total = N * MDIM;
    const int m = *m_ptr;
    if (idx < total) {
        if ((idx & 63) >= m) out[idx] = 0.0f;
    } else if (idx < total + N) {
        out[idx] = 1.0f;
    }
}

extern "C" void kernel_launch(void* const* d_in, const int* in_sizes, int n_in,
                              void* d_out, int out_size, void* d_ws, size_t ws_size,
                              hipStream_t stream) {
    const int*   batch_idx = (const int*)d_in[0];
    const float* locs      = (const float*)d_in[1];
    const int*   nn        = (const int*)d_in[2];
    const float* log_ls    = (const float*)d_in[3];
    float* out = (float*)d_out;
    const int N = in_sizes[0];
    int* m_ptr = (int*)d_ws;

    int nblocks = (N < 2048) ? N : 2048;
    vecchia_solve_kernel<<<nblocks, 32, 0, stream>>>(batch_idx, locs, nn, log_ls, out, N);
    vecchia_reduce_kernel<<<1, 1024, 0, stream>>>(out, batch_idx, N, m_ptr);
    const int total = N * MDIM + N;
    vecchia_mask_kernel<<<(total + 255) / 256, 256, 0, stream>>>(out, m_ptr, N);
}